// ClusterAttn_20753281974736
// MI455X (gfx1250) — compile-verified
//
#include <hip/hip_runtime.h>
#include <math.h>

// ---------------------------------------------------------------------------
// ClusterAttn for MI455X (gfx1250): fp32 end-to-end using V_WMMA_F32_16X16X4_F32
// plus GLOBAL_LOAD_ASYNC_TO_LDS_B128 staging (ASYNCcnt) for non-transposed tiles.
// B=2, H=W=128, C=256, nh=8, hd=32, pn=16 -> S=256 patches of P=64 pixels,
// L=16384. Paired batches: q=[0,1,0,1], ctx=[0,1,1,0].
// ---------------------------------------------------------------------------

typedef __attribute__((ext_vector_type(2))) float v2f;
typedef __attribute__((ext_vector_type(8))) float v8f;

#define WMMA_F32(a, b, c) \
  __builtin_amdgcn_wmma_f32_16x16x4_f32(false, (a), false, (b), (short)0, (c), false, false)

#define EPSV 1e-6f
#define V8F_ZERO ((v8f){0.f, 0.f, 0.f, 0.f, 0.f, 0.f, 0.f, 0.f})

// Async 16B global -> LDS copy (per-lane addresses). Tracked by ASYNCcnt.
static __device__ __forceinline__ void async_copy_f4(const float* g, float* l) {
  const unsigned lds = (unsigned)(size_t)l;   // low 32 bits of flat addr = LDS offset
  asm volatile("global_load_async_to_lds_b128 %0, %1, off" :: "v"(lds), "v"(g) : "memory");
}
static __device__ __forceinline__ void wait_async0() {
  asm volatile("s_wait_asynccnt 0" ::: "memory");
}

static __device__ __forceinline__ float wave_reduce_sum(float v) {
  #pragma unroll
  for (int off = 16; off > 0; off >>= 1) v += __shfl_xor(v, off, 32);
  return v;
}

// ---------------------------------------------------------------------------
// Generic Out[M,N] = A[M,256] * W[N,256]^T + bias[N]
// block = 128 threads (4 waves), tile 64(M) x 128(N), each wave 32x64 (2x4 frags).
// A fragment: lane = m + 16*(k>>1), reg = k&1
// B fragment: lane = n + 16*(k>>1), reg = k&1
// C fragment: reg = m&7, laneHalf = m>>3, lane = n
// As pad = 36 floats (144 B: 16B-aligned rows for async B128, distinct banks).
// ---------------------------------------------------------------------------
__global__ __launch_bounds__(128) void gemm_nt_bias(
    const float* __restrict__ A, const float* __restrict__ W,
    const float* __restrict__ bias, float* __restrict__ Out, int N) {
  __shared__ __align__(16) float As[64][36];    // [m][k]
  __shared__ __align__(16) float Bs[32][129];   // [k][n] (transposed stores)

  const int tid   = threadIdx.x;
  const int lane  = tid & 31;
  const int wave  = tid >> 5;
  const int lmod  = lane & 15;
  const int lhalf = lane >> 4;
  const int wm = (wave >> 1) * 32;   // wave M offset in tile
  const int wn = (wave & 1) * 64;    // wave N offset in tile
  const long m0 = (long)blockIdx.y * 64;
  const long n0 = (long)blockIdx.x * 128;

  v8f acc[2][4];
  #pragma unroll
  for (int i = 0; i < 2; ++i)
    #pragma unroll
    for (int j = 0; j < 4; ++j) acc[i][j] = V8F_ZERO;

  for (int k0 = 0; k0 < 256; k0 += 32) {
    __syncthreads();   // previous chunk fully consumed
    // A tile 64x32: async 16B chunks straight into LDS (no VGPR round-trip)
    #pragma unroll
    for (int i = tid; i < 512; i += 128) {
      const int r = i >> 3, c = (i & 7) * 4;
      async_copy_f4(A + (m0 + r) * 256 + k0 + c, &As[r][c]);
    }
    // W tile 128x32 transposed: Bs[k][n] = W[n0+n][k0+k] (needs VGPR transpose)
    #pragma unroll
    for (int i = tid; i < 1024; i += 128) {
      const int r = i >> 3, c = (i & 7) * 4;
      const float4 v4 = *(const float4*)(W + (n0 + r) * 256 + k0 + c);
      Bs[c + 0][r] = v4.x; Bs[c + 1][r] = v4.y;
      Bs[c + 2][r] = v4.z; Bs[c + 3][r] = v4.w;
    }
    wait_async0();
    __syncthreads();
    #pragma unroll
    for (int kk = 0; kk < 32; kk += 4) {
      const int ka = kk + 2 * lhalf;
      v2f a[2], b[4];
      #pragma unroll
      for (int f = 0; f < 2; ++f) {
        const int row = wm + f * 16 + lmod;
        a[f].x = As[row][ka];
        a[f].y = As[row][ka + 1];
      }
      #pragma unroll
      for (int f = 0; f < 4; ++f) {
        const int col = wn + f * 16 + lmod;
        b[f].x = Bs[ka][col];
        b[f].y = Bs[ka + 1][col];
      }
      #pragma unroll
      for (int fm = 0; fm < 2; ++fm)
        #pragma unroll
        for (int fn = 0; fn < 4; ++fn)
          acc[fm][fn] = WMMA_F32(a[fm], b[fn], acc[fm][fn]);
    }
  }

  #pragma unroll
  for (int fm = 0; fm < 2; ++fm) {
    #pragma unroll
    for (int fn = 0; fn < 4; ++fn) {
      const long col = n0 + wn + fn * 16 + lmod;
      const float bv = bias[col];
      #pragma unroll
      for (int r = 0; r < 8; ++r) {
        const long row = m0 + wm + fm * 16 + 8 * lhalf + r;
        Out[row * N + col] = acc[fm][fn][r] + bv;
      }
    }
  }
}

// ---------------------------------------------------------------------------
// Cluster pooling: per (b,s) compute fg/bg sums w/ EPS rules, then pooled
// c_k = l2n(sum fg*k_n) + l2n(sum bg*k_n), c_v = sum(fg*v)/fs + sum(bg*v)/bs.
// k is L2-normalized per pixel first. block = 256 thr: h = tid>>5, d = tid&31.
// ---------------------------------------------------------------------------
__global__ __launch_bounds__(256) void cluster_pool(
    const float* __restrict__ qkv, const float* __restrict__ mask,
    float* __restrict__ ck, float* __restrict__ cv) {
  __shared__ float Ms[64];
  const int s = blockIdx.x;          // patch 0..255  (s = sy*16 + sx)
  const int b = blockIdx.y;          // 0..1
  const int h = threadIdx.x >> 5;
  const int d = threadIdx.x & 31;
  const int sy = s >> 4, sx = s & 15;

  if (threadIdx.x < 64) {
    const int iy = threadIdx.x >> 3, ix = threadIdx.x & 7;
    Ms[threadIdx.x] = mask[((size_t)b * 128 + sy * 8 + iy) * 128 + sx * 8 + ix];
  }
  __syncthreads();

  float fgsum = 0.f;
  #pragma unroll
  for (int p = 0; p < 64; ++p) fgsum += Ms[p];
  const float bgsum = 64.f - fgsum;
  float fs  = fgsum < EPSV ? EPSV : fgsum;
  float bs2 = bgsum < EPSV ? EPSV : bgsum;
  fs  = (fs < bs2)  ? EPSV : fs;    // in-place order: bs2 test uses updated fs
  bs2 = (fs >= bs2) ? EPSV : bs2;
  const bool fgZ = fs  < 1.f;
  const bool bgZ = bs2 < 1.f;

  float fk = 0.f, fv = 0.f, bk = 0.f, bv = 0.f;
  for (int p = 0; p < 64; ++p) {
    const int y = sy * 8 + (p >> 3), x = sx * 8 + (p & 7);
    const size_t base = ((size_t)b * 16384 + (size_t)y * 128 + x) * 768;
    const float kval = qkv[base + 256 + h * 32 + d];
    const float vval = qkv[base + 512 + h * 32 + d];
    const float nsq  = wave_reduce_sum(kval * kval);  // per-pixel ||k||^2 over hd
    const float kn   = kval / fmaxf(sqrtf(nsq), 1e-12f);
    const float m  = Ms[p];
    const float wf = fgZ ? 0.f : m;
    const float wb = bgZ ? 0.f : (1.f - m);
    fk += wf * kn; fv += wf * vval;
    bk += wb * kn; bv += wb * vval;
  }
  const float nf = fmaxf(sqrtf(wave_reduce_sum(fk * fk)), 1e-12f);
  const float nb = fmaxf(sqrtf(wave_reduce_sum(bk * bk)), 1e-12f);
  const size_t o = (((size_t)b * 8 + h) * 256 + s) * 32 + d;
  ck[o] = fk / nf + bk / nb;
  cv[o] = fv / fs + bv / bs2;
}

// ---------------------------------------------------------------------------
// Attention: per (bb, h): logits = Qn[64,32] @ CK^T[32,256] via WMMA,
// softmax over S=256 in LDS, out = P @ CV[256,32] via WMMA.
// block = 128 thr (4 waves), 16 L-rows per wave. CK/CV/Q staged via async B128.
// ---------------------------------------------------------------------------
__global__ __launch_bounds__(128) void attn_kernel(
    const float* __restrict__ qkv, const float* __restrict__ ck,
    const float* __restrict__ cv, const float* __restrict__ scale_p,
    float* __restrict__ obuf) {
  __shared__ __align__(16) float CKs[256][36];   // [s][d]
  __shared__ __align__(16) float CVs[256][36];   // [s][d]
  __shared__ __align__(16) float Qs[64][36];     // [row][d], normalized in place
  __shared__ __align__(16) float Ls[4][16][257]; // per-wave logits / probabilities

  const int bb = blockIdx.z;                      // 0..3
  const int h  = blockIdx.y;                      // 0..7
  const int l0 = blockIdx.x * 64;
  const int qb = bb & 1;                          // q pairing [0,1,0,1]
  const int cb = (bb == 1 || bb == 2) ? 1 : 0;    // ctx pairing [0,1,1,0]
  const int tid   = threadIdx.x;
  const int lane  = tid & 31;
  const int w     = tid >> 5;
  const int lmod  = lane & 15;
  const int lhalf = lane >> 4;

  // stage CK/CV (256x32 each) and raw Q (64x32) via async B128 copies
  const size_t ctx_base = (((size_t)cb * 8 + h) * 256) * 32;
  #pragma unroll
  for (int i = tid; i < 2048; i += 128) {      // 256 rows x 8 chunks
    const int ss = i >> 3, c = (i & 7) * 4;
    async_copy_f4(ck + ctx_base + (size_t)ss * 32 + c, &CKs[ss][c]);
    async_copy_f4(cv + ctx_base + (size_t)ss * 32 + c, &CVs[ss][c]);
  }
  #pragma unroll
  for (int i = tid; i < 512; i += 128) {       // 64 rows x 8 chunks
    const int r = i >> 3, c = (i & 7) * 4;
    async_copy_f4(qkv + ((size_t)qb * 16384 + l0 + r) * 768 + h * 32 + c, &Qs[r][c]);
  }
  wait_async0();
  __syncthreads();

  // q <- l2norm(q) * scale * hd^-0.5
  const float qscale = scale_p[0] * 0.17677669529663689f;
  for (int r = tid; r < 64; r += 128) {
    float ssq = 0.f;
    #pragma unroll
    for (int dd = 0; dd < 32; ++dd) ssq += Qs[r][dd] * Qs[r][dd];
    const float sc = qscale / fmaxf(sqrtf(ssq), 1e-12f);
    #pragma unroll
    for (int dd = 0; dd < 32; ++dd) Qs[r][dd] *= sc;
  }
  __syncthreads();

  // preload the 8 k-step Q fragments (invariant across all 16 S-tiles)
  v2f qa[8];
  #pragma unroll
  for (int k8 = 0; k8 < 8; ++k8) {
    const int ka = k8 * 4 + 2 * lhalf;
    qa[k8].x = Qs[w * 16 + lmod][ka];
    qa[k8].y = Qs[w * 16 + lmod][ka + 1];
  }

  // logits: 16 rows x 256 cols per wave (16 tiles x 8 k-steps of WMMA)
  for (int nt = 0; nt < 16; ++nt) {
    v8f acc = V8F_ZERO;
    #pragma unroll
    for (int k8 = 0; k8 < 8; ++k8) {
      const int ka = k8 * 4 + 2 * lhalf;
      v2f bf;
      bf.x = CKs[nt * 16 + lmod][ka];      // B(k=d, n=s) = CK[s][d]
      bf.y = CKs[nt * 16 + lmod][ka + 1];
      acc = WMMA_F32(qa[k8], bf, acc);
    }
    #pragma unroll
    for (int r = 0; r < 8; ++r)
      Ls[w][8 * lhalf + r][nt * 16 + lmod] = acc[r];
  }
  __syncthreads();

  // softmax over S=256: lanes 0..15 of each wave each own one row
  if (lane < 16) {
    float* row = &Ls[w][lane][0];
    float mx = -3.402823466e38f;
    for (int s2 = 0; s2 < 256; ++s2) mx = fmaxf(mx, row[s2]);
    float sum = 0.f;
    for (int s2 = 0; s2 < 256; ++s2) {
      const float e = __expf(row[s2] - mx);
      row[s2] = e;
      sum += e;
    }
    const float inv = 1.f / sum;
    for (int s2 = 0; s2 < 256; ++s2) row[s2] *= inv;
  }
  __syncthreads();

  // out = P[16,256] @ CV[256,32] per wave; kk outer so A frag feeds both N-tiles
  v8f acc2[2] = {V8F_ZERO, V8F_ZERO};
  for (int kk = 0; kk < 256; kk += 4) {
    const int ka = kk + 2 * lhalf;
    v2f a;
    a.x = Ls[w][lmod][ka];
    a.y = Ls[w][lmod][ka + 1];
    #pragma unroll
    for (int nt2 = 0; nt2 < 2; ++nt2) {
      v2f bf;
      bf.x = CVs[ka][nt2 * 16 + lmod];     // B(k=s, n=d) = CV[s][d]
      bf.y = CVs[ka + 1][nt2 * 16 + lmod];
      acc2[nt2] = WMMA_F32(a, bf, acc2[nt2]);
    }
  }
  #pragma unroll
  for (int nt2 = 0; nt2 < 2; ++nt2) {
    #pragma unroll
    for (int r = 0; r < 8; ++r) {
      const size_t row = (size_t)bb * 16384 + l0 + w * 16 + 8 * lhalf + r;
      obuf[row * 256 + h * 32 + nt2 * 16 + lmod] = acc2[nt2][r];
    }
  }
}

// ---------------------------------------------------------------------------
extern "C" void kernel_launch(void* const* d_in, const int* in_sizes, int n_in,
                              void* d_out, int out_size, void* d_ws, size_t ws_size,
                              hipStream_t stream) {
  (void)in_sizes; (void)n_in; (void)out_size; (void)ws_size;
  const float* x      = (const float*)d_in[0];   // [2,128,128,256]
  const float* mask   = (const float*)d_in[1];   // [2,1,128,128]
  const float* qkv_w  = (const float*)d_in[2];   // [768,256]
  const float* qkv_b  = (const float*)d_in[3];   // [768]
  const float* proj_w = (const float*)d_in[4];   // [256,256]
  const float* proj_b = (const float*)d_in[5];   // [256]
  const float* scale  = (const float*)d_in[6];   // [1]
  float* out = (float*)d_out;                    // [4,128,128,256]

  float* ws   = (float*)d_ws;
  float* qkv  = ws;                                   // [2,16384,768]
  float* ck   = qkv + (size_t)2 * 16384 * 768;        // [2,8,256,32]
  float* cv   = ck + (size_t)2 * 8 * 256 * 32;        // [2,8,256,32]
  float* obuf = cv + (size_t)2 * 8 * 256 * 32;        // [4,16384,256]

  // 1) qkv = x @ qkv_w^T + qkv_b   (M=32768, N=768, K=256)
  gemm_nt_bias<<<dim3(768 / 128, 32768 / 64), 128, 0, stream>>>(x, qkv_w, qkv_b, qkv, 768);

  // 2) fg/bg cluster pooling -> c_k, c_v
  cluster_pool<<<dim3(256, 2), 256, 0, stream>>>(qkv, mask, ck, cv);

  // 3) attention (QK^T, softmax, PV) -> obuf [4, L, C]
  attn_kernel<<<dim3(16384 / 64, 8, 4), 128, 0, stream>>>(qkv, ck, cv, scale, obuf);

  // 4) out = obuf @ proj_w^T + proj_b   (M=65536, N=256, K=256)
  gemm_nt_bias<<<dim3(256 / 128, 65536 / 64), 128, 0, stream>>>(obuf, proj_w, proj_b, out, 256);
}